// NeighborhoodAttentionModule_6923487282189
// MI455X (gfx1250) — compile-verified
//
#include <hip/hip_runtime.h>
#include <math.h>

typedef __attribute__((ext_vector_type(2))) float v2f;
typedef __attribute__((ext_vector_type(4))) float f4;
typedef __attribute__((ext_vector_type(8))) float v8f;

#define D_GNN 256
#define KNBR  16
#define ATTN  64
#define TB    32              // batch rows per block (2 M-tiles of 16)
#define GPAD  (2 * D_GNN + 2) // padded row stride: mod 64 == 2 -> conflict-free column reads
#define QPAD  (ATTN + 2)
#define EPSV  1e-6f

// fused neighborhood attention: one block = 32 batch rows, 256 threads = 8 waves (wave32)
__launch_bounds__(256)
__global__ void neigh_attn_fused(const float* __restrict__ center,
                                 const float* __restrict__ embs,
                                 const int*   __restrict__ nidx,
                                 const float* __restrict__ nwt,
                                 const float* __restrict__ Wq,
                                 const float* __restrict__ Wk,
                                 const float* __restrict__ Wg,
                                 const float* __restrict__ bg,
                                 float* __restrict__ out,
                                 int nrows)
{
    // sGate row = [ center(256) | qk -> later weighted(256) | pad(2) ]
    // qk lives in the second half until the scores phase ends; the weighted
    // sum then overwrites it (phases separated by barriers).
    __shared__ float sGate[TB][GPAD];
    __shared__ float sQ[TB][QPAD];
    __shared__ float sS[TB][KNBR];
    __shared__ float sA[TB][KNBR];
    __shared__ int   sI[TB][KNBR];

    const int t     = threadIdx.x;
    const int wid   = t >> 5;          // wave id 0..7
    const int ln    = t & 31;          // lane
    const int lhalf = ln >> 4;         // 0: lanes 0-15, 1: lanes 16-31
    const int lmod  = ln & 15;
    const int row0  = blockIdx.x * TB;

    // ---- stage center tile into LDS (first half of gate input) ----
    for (int i = t; i < TB * (D_GNN / 4); i += 256) {
        const int r = i >> 6;
        const int c = (i & 63) * 4;
        f4 v = *(const f4*)(center + (size_t)(row0 + r) * D_GNN + c);
        sGate[r][c + 0] = v.x; sGate[r][c + 1] = v.y;
        sGate[r][c + 2] = v.z; sGate[r][c + 3] = v.w;
    }
    __syncthreads();

    // ---- q = C @ Wq : 32x64 ; 8 waves = 2 M-tiles x 4 N-tiles, K=256 ----
    {
        const int mt = wid >> 2;           // M-tile 0/1
        const int nt = wid & 3;            // N-tile 0..3
        const int n  = nt * 16 + lmod;
        const int mr = mt * 16 + lmod;     // A row for this lane
        v8f c = {};
        for (int s = 0; s < D_GNN / 4; ++s) {
            const int kb = 4 * s + 2 * lhalf;   // A 16x4: lanes 0-15 K={0,1}, 16-31 K={2,3}
            v2f a, b;
            a.x = sGate[mr][kb];
            a.y = sGate[mr][kb + 1];
            b.x = Wq[(size_t)kb * ATTN + n];    // B 4x16: row k striped across lanes (N)
            b.y = Wq[(size_t)(kb + 1) * ATTN + n];
            c = __builtin_amdgcn_wmma_f32_16x16x4_f32(false, a, false, b,
                                                      (short)0, c, false, false);
        }
        for (int v = 0; v < 8; ++v)        // D: VGPR v -> M = v + 8*lhalf
            sQ[mt * 16 + v + 8 * lhalf][n] = c[v];
    }
    __syncthreads();

    // ---- qk = q @ Wk^T : 32x256 ; B fragment shared across both M-tiles ----
    for (int jj = 0; jj < 2; ++jj) {
        const int j = 2 * wid + jj;        // N-tile 0..15
        const int n = j * 16 + lmod;
        v8f c0 = {}, c1 = {};
        for (int s = 0; s < ATTN / 4; ++s) {
            const int kb = 4 * s + 2 * lhalf;
            v2f a0, a1, b;
            b.x  = Wk[(size_t)n * ATTN + kb];   // Wk^T[k][n] = Wk[n][k], Wk is [256][64]
            b.y  = Wk[(size_t)n * ATTN + kb + 1];
            a0.x = sQ[lmod][kb];      a0.y = sQ[lmod][kb + 1];
            a1.x = sQ[16 + lmod][kb]; a1.y = sQ[16 + lmod][kb + 1];
            c0 = __builtin_amdgcn_wmma_f32_16x16x4_f32(false, a0, false, b,
                                                       (short)0, c0, false, false);
            c1 = __builtin_amdgcn_wmma_f32_16x16x4_f32(false, a1, false, b,
                                                       (short)0, c1, false, false);
        }
        for (int v = 0; v < 8; ++v) {
            sGate[v + 8 * lhalf][D_GNN + n]      = c0[v];   // qk aliased into 2nd half
            sGate[16 + v + 8 * lhalf][D_GNN + n] = c1[v];
        }
    }
    __syncthreads();

    // ---- scores: (row, neighbor) pairs; 256-wide dot qk . emb[idx] (L2-hot) ----
    for (int pp = t; pp < TB * KNBR; pp += 256) {
        const int r  = pp >> 4;
        const int nn = pp & 15;
        const int gi = (row0 + r) * KNBR + nn;
        const int id = nidx[gi];
        const float w = nwt[gi];
        sI[r][nn] = id;
        const f4* e = (const f4*)(embs + (size_t)id * D_GNN);
        float dot = 0.f;
        for (int j = 0; j < D_GNN / 4; ++j) {
            f4 v = e[j];
            dot += v.x * sGate[r][D_GNN + 4 * j + 0] + v.y * sGate[r][D_GNN + 4 * j + 1]
                 + v.z * sGate[r][D_GNN + 4 * j + 2] + v.w * sGate[r][D_GNN + 4 * j + 3];
        }
        float s = dot * 0.125f + __logf(fmaxf(w, EPSV));  // ATTN^-0.5 = 1/8
        if (w < EPSV) s = -INFINITY;
        sS[r][nn] = s;
    }
    __syncthreads();

    // ---- softmax over 16 neighbors (all-padded rows -> 0, matches nan_to_num) ----
    for (int pp = t; pp < TB * KNBR; pp += 256) {
        const int r  = pp >> 4;
        const int nn = pp & 15;
        float mx = -INFINITY;
        for (int j = 0; j < KNBR; ++j) mx = fmaxf(mx, sS[r][j]);
        float a = 0.f;
        if (mx != -INFINITY) {
            float sum = 0.f;
            for (int j = 0; j < KNBR; ++j) sum += __expf(sS[r][j] - mx);
            a = __expf(sS[r][nn] - mx) / sum;
        }
        sA[r][nn] = a;
    }
    __syncthreads();

    // ---- weighted neighbor sum -> overwrite qk region with weighted(256) ----
    for (int pp = t; pp < TB * 16; pp += 256) {
        const int r  = pp >> 4;
        const int c0 = (pp & 15) * 16;     // 16 contiguous columns per work item
        f4 acc[4] = {};
        for (int nn = 0; nn < KNBR; ++nn) {
            const float a = sA[r][nn];
            const f4* p = (const f4*)(embs + (size_t)sI[r][nn] * D_GNN + c0);
            for (int j = 0; j < 4; ++j) {
                f4 v = p[j];
                acc[j].x += a * v.x; acc[j].y += a * v.y;
                acc[j].z += a * v.z; acc[j].w += a * v.w;
            }
        }
        for (int j = 0; j < 4; ++j) {
            sGate[r][D_GNN + c0 + 4 * j + 0] = acc[j].x;
            sGate[r][D_GNN + c0 + 4 * j + 1] = acc[j].y;
            sGate[r][D_GNN + c0 + 4 * j + 2] = acc[j].z;
            sGate[r][D_GNN + c0 + 4 * j + 3] = acc[j].w;
        }
    }
    __syncthreads();

    // ---- gate GEMM [32,512]@[512,256]: each Wg fragment feeds 2 M-tiles ----
    for (int jj = 0; jj < 2; ++jj) {
        const int j = 2 * wid + jj;        // N-tile 0..15
        const int n = j * 16 + lmod;
        v8f c0 = {}, c1 = {};
        for (int s = 0; s < (2 * D_GNN) / 4; ++s) {
            const int kb = 4 * s + 2 * lhalf;
            v2f a0, a1, b;
            b.x  = Wg[(size_t)kb * D_GNN + n];
            b.y  = Wg[(size_t)(kb + 1) * D_GNN + n];
            a0.x = sGate[lmod][kb];      a0.y = sGate[lmod][kb + 1];
            a1.x = sGate[16 + lmod][kb]; a1.y = sGate[16 + lmod][kb + 1];
            c0 = __builtin_amdgcn_wmma_f32_16x16x4_f32(false, a0, false, b,
                                                       (short)0, c0, false, false);
            c1 = __builtin_amdgcn_wmma_f32_16x16x4_f32(false, a1, false, b,
                                                       (short)0, c1, false, false);
        }
        const float bias = bg[n];
        for (int v = 0; v < 8; ++v) {
            const int m0 = v + 8 * lhalf;
            const int m1 = 16 + v + 8 * lhalf;
            if (row0 + m0 < nrows) {
                const float g = 1.f / (1.f + __expf(-(c0[v] + bias)));
                out[(size_t)(row0 + m0) * D_GNN + n] =
                    g * sGate[m0][n] + (1.f - g) * sGate[m0][D_GNN + n];
            }
            if (row0 + m1 < nrows) {
                const float g = 1.f / (1.f + __expf(-(c1[v] + bias)));
                out[(size_t)(row0 + m1) * D_GNN + n] =
                    g * sGate[m1][n] + (1.f - g) * sGate[m1][D_GNN + n];
            }
        }
    }
}

extern "C" void kernel_launch(void* const* d_in, const int* in_sizes, int n_in,
                              void* d_out, int out_size, void* d_ws, size_t ws_size,
                              hipStream_t stream) {
    const float* center = (const float*)d_in[0];
    const float* embs   = (const float*)d_in[1];
    const int*   nidx   = (const int*)d_in[2];
    const float* nwt    = (const float*)d_in[3];
    const float* Wq     = (const float*)d_in[4];
    const float* Wk     = (const float*)d_in[5];
    const float* Wg     = (const float*)d_in[6];
    const float* bg     = (const float*)d_in[7];
    float* out = (float*)d_out;

    const int B = in_sizes[0] / D_GNN;          // 65536
    dim3 grid((B + TB - 1) / TB), block(256);
    hipLaunchKernelGGL(neigh_attn_fused, grid, block, 0, stream,
                       center, embs, nidx, nwt, Wq, Wk, Wg, bg, out, B);
}